// HalfEdgeMeshPool_7662221656641
// MI455X (gfx1250) — compile-verified
//
#include <hip/hip_runtime.h>
#include <stdint.h>

// Native vector type (required by __builtin_nontemporal_load/store; HIP's
// float4 is a struct and is rejected).
typedef float  vf4 __attribute__((ext_vector_type(4)));
typedef int    vi4 __attribute__((ext_vector_type(4)));

// Problem constants (from the reference): B meshes, C channels, N half-edges,
// T surviving groups.
namespace {
constexpr int B = 8;
constexpr int C = 256;
constexpr int N = 40000;   // divisible by 4 -> int4/float4/b128 tiles stay in-bounds
constexpr int T = 20000;   // divisible by 4 -> float4 rows in scale kernel

constexpr int TILE_N  = 1024;          // half-edges per block tile
constexpr int NT      = 4;             // half-edges per thread (one b128 gid load)
constexpr int THREADS = TILE_N / NT;   // 256 threads = 8 waves (wave32)
constexpr int CCHUNK  = 64;            // channels per block (grid.z = C/CCHUNK)
}

// ---------------------------------------------------------------------------
// Pass 1: per-group element counts. int4-vectorized gid loads, 4 global f32
// atomics per thread; counts (640 KB) are trivially L2-resident.
// ---------------------------------------------------------------------------
__global__ void heps_count_kernel(const int* __restrict__ gid,
                                  float* __restrict__ counts) {
  unsigned i4 = blockIdx.x * blockDim.x + threadIdx.x;
  constexpr unsigned TOTAL4 = (unsigned)(B * N) / 4;      // 80,000
  if (i4 < TOTAL4) {
    unsigned i = i4 * 4u;
    unsigned b = i / (unsigned)N;                         // N % 4 == 0: same batch
    vi4 g = reinterpret_cast<const vi4*>(gid)[i4];
    float* cb = counts + b * (unsigned)T;
    atomicAdd(cb + g.x, 1.0f);
    atomicAdd(cb + g.y, 1.0f);
    atomicAdd(cb + g.z, 1.0f);
    atomicAdd(cb + g.w, 1.0f);
  }
}

// ---------------------------------------------------------------------------
// Pass 2: counts -> 1 / max(count, 1)  (matches reference's maximum(counts,1)).
// ---------------------------------------------------------------------------
__global__ void heps_recip_kernel(float* __restrict__ counts) {
  unsigned i = blockIdx.x * blockDim.x + threadIdx.x;
  if (i < (unsigned)(B * T)) {
    counts[i] = 1.0f / fmaxf(counts[i], 1.0f);
  }
}

// ---------------------------------------------------------------------------
// Pass 3: main scatter-add. Each block:
//   * async-copies its 1024-entry gid tile into LDS (CDNA5 async engine,
//     ASYNCcnt) and waits with s_wait_asynccnt,
//   * each thread owns 4 consecutive half-edges, loops over a 64-channel
//     chunk doing one coalesced NON-TEMPORAL 16B vector load + 4 atomic adds.
// Cache policy: the 163.8 MB accumulator stays resident in the 192 MB L2
// (default-RT atomics); the 327 MB feature stream is marked NT so it does
// not evict the accumulator.
// ---------------------------------------------------------------------------
__global__ void heps_scatter_kernel(const float* __restrict__ feat,
                                    const int* __restrict__ gid,
                                    float* __restrict__ out) {
  __shared__ int tile[TILE_N];

  const int b  = blockIdx.y;
  const int n0 = blockIdx.x * TILE_N;
  const int c0 = blockIdx.z * CCHUNK;
  const int t  = threadIdx.x;
  const int n  = n0 + NT * t;                 // N % 4 == 0 => n valid => n..n+3 valid
  const bool active = (n < N);

  if (active) {
    // Low 32 bits of a generic LDS pointer == workgroup-relative LDS byte
    // address (ISA flat-aperture rule) -> VDST of the async load.
    uint32_t lds_addr = (uint32_t)(uintptr_t)(&tile[NT * t]);
    uint32_t voff     = (uint32_t)(((size_t)b * N + (size_t)n) * sizeof(int));
    unsigned long long sbase = (unsigned long long)(uintptr_t)gid;
    asm volatile("global_load_async_to_lds_b128 %0, %1, %2 offset:0"
                 :
                 : "v"(lds_addr), "v"(voff), "s"(sbase)
                 : "memory");
  }
  asm volatile("s_wait_asynccnt 0x0" ::: "memory");

  if (!active) return;

  const int g0 = tile[NT * t + 0];
  const int g1 = tile[NT * t + 1];
  const int g2 = tile[NT * t + 2];
  const int g3 = tile[NT * t + 3];

  const float* frow  = feat + (size_t)b * C * N + (size_t)c0 * N + n;
  float*       obase = out  + (size_t)b * C * T + (size_t)c0 * T;

#pragma unroll 4
  for (int c = 0; c < CCHUNK; ++c) {
    // Coalesced, 16B-aligned, non-temporal (evict-first) streaming load.
    const vf4 v = __builtin_nontemporal_load(
        reinterpret_cast<const vf4*>(frow + (size_t)c * N));
    float* orow = obase + (size_t)c * T;
    atomicAdd(orow + g0, v.x);
    atomicAdd(orow + g1, v.y);
    atomicAdd(orow + g2, v.z);
    atomicAdd(orow + g3, v.w);
  }
}

// ---------------------------------------------------------------------------
// Pass 4: out *= invcount (broadcast over channels), float4 vectorized,
// 32-bit index math. Reads hit the L2-resident accumulator; final stores are
// non-temporal (result is never re-read on device).
// ---------------------------------------------------------------------------
__global__ void heps_scale_kernel(float* __restrict__ out,
                                  const float* __restrict__ invcnt) {
  constexpr unsigned TOTAL4 = (unsigned)((size_t)B * C * T / 4);  // 10,240,000
  unsigned i4 = blockIdx.x * blockDim.x + threadIdx.x;
  if (i4 >= TOTAL4) return;
  unsigned i  = i4 * 4u;                                          // < 2^31
  unsigned b  = i / (unsigned)(C * T);
  unsigned tt = i % (unsigned)T;                                  // T % 4 == 0
  vf4 v        = reinterpret_cast<vf4*>(out)[i4];
  const vf4 ic = *reinterpret_cast<const vf4*>(invcnt + b * (unsigned)T + tt);
  v.x *= ic.x; v.y *= ic.y; v.z *= ic.z; v.w *= ic.w;
  __builtin_nontemporal_store(v, reinterpret_cast<vf4*>(out) + i4);
}

// ---------------------------------------------------------------------------
extern "C" void kernel_launch(void* const* d_in, const int* in_sizes, int n_in,
                              void* d_out, int out_size, void* d_ws, size_t ws_size,
                              hipStream_t stream) {
  (void)in_sizes; (void)n_in; (void)out_size; (void)ws_size;

  const float* feat = (const float*)d_in[0];   // [B, C, N] float32
  const int*   gid  = (const int*)d_in[1];     // [B, N]    int32
  float* out    = (float*)d_out;               // [B, C, T] float32
  float* counts = (float*)d_ws;                // [B, T]    scratch (640 KB)

  // Zero the accumulators (graph-capturable async memsets).
  (void)hipMemsetAsync(out,    0, (size_t)B * C * T * sizeof(float), stream);
  (void)hipMemsetAsync(counts, 0, (size_t)B * T * sizeof(float),     stream);

  heps_count_kernel<<<((B * N / 4) + 255) / 256, 256, 0, stream>>>(gid, counts);
  heps_recip_kernel<<<(B * T + 255) / 256, 256, 0, stream>>>(counts);

  dim3 grid((N + TILE_N - 1) / TILE_N, B, C / CCHUNK);   // (40, 8, 4) = 1280 blocks
  heps_scatter_kernel<<<grid, THREADS, 0, stream>>>(feat, gid, out);

  unsigned total4 = (unsigned)((size_t)B * C * T / 4);
  heps_scale_kernel<<<(total4 + 255) / 256, 256, 0, stream>>>(out, counts);
}